// FFN_pairwise_z_61186104099111
// MI455X (gfx1250) — compile-verified
//
#include <hip/hip_runtime.h>

typedef __attribute__((ext_vector_type(2))) float v2f;
typedef __attribute__((ext_vector_type(8))) float v8f;

#define ZD 128
#define NN 1024
#define QROW 33   // float4 per LDS tile row: 132 floats, 16B-aligned rows, 132%64=4 -> distinct banks

// ---------------------------------------------------------------------------
// out[i,h] = sum_z X[i,z] * W[z,h]  (+ bias[h] if bias != nullptr)
// One wave computes one 16x16 tile via V_WMMA_F32_16X16X4_F32 (K-loop of 4).
// Fragment layouts per CDNA5 ISA 7.12.2.
// ---------------------------------------------------------------------------
__global__ __launch_bounds__(32) void gemm16_wmma(const float* __restrict__ X,
                                                  const float* __restrict__ W,
                                                  const float* __restrict__ bias,
                                                  float* __restrict__ out) {
  const int tm   = blockIdx.x;       // 0..63 : 16-row tile of X
  const int tn   = blockIdx.y;       // 0..7  : 16-col tile of output
  const int lane = threadIdx.x;      // 0..31 (wave32)
  const int half = lane >> 4;
  const int lr   = lane & 15;

  const float* xrow = X + (tm * 16 + lr) * ZD;   // row M=lr of A tile
  const float* wcol = W + tn * 16 + lr;          // col N=lr of B tile

  v8f acc = {};
#pragma unroll 4
  for (int k = 0; k < ZD; k += 4) {
    v2f av, bv;
    const int ka = k + 2 * half;
    av.x = xrow[ka + 0];
    av.y = xrow[ka + 1];
    bv.x = wcol[(ka + 0) * ZD];
    bv.y = wcol[(ka + 1) * ZD];
    acc = __builtin_amdgcn_wmma_f32_16x16x4_f32(
        /*neg_a=*/false, av, /*neg_b=*/false, bv,
        /*c_mod=*/(short)0, acc, /*reuse_a=*/false, /*reuse_b=*/false);
  }

  const float badd = bias ? bias[tn * 16 + lr] : 0.0f;
#pragma unroll
  for (int r = 0; r < 8; ++r) {
    out[(tm * 16 + r + 8 * half) * ZD + tn * 16 + lr] = acc[r] + badd;
  }
}

// ---------------------------------------------------------------------------
// score[i,j] = relu( sum_h relu(a[i,h] + c[j,h]) * W2[h] + b2 )
// Block = 256 threads = one 16(i) x 16(j) tile, LDS-staged as float4.
// Deterministic output: per-block row sums -> OPpart[jb][i], col sums -> COpart[ib][j].
// ---------------------------------------------------------------------------
__global__ __launch_bounds__(256) void pairwise_score(const float4* __restrict__ A4,
                                                      const float4* __restrict__ C4,
                                                      const float4* __restrict__ W24,
                                                      const float* __restrict__ b2,
                                                      float* __restrict__ OPpart,
                                                      float* __restrict__ COpart) {
  __shared__ float4 sa[16 * QROW];
  __shared__ float4 sc[16 * QROW];
  __shared__ float4 sw[ZD / 4];
  __shared__ float  ss[16][17];

  const int tid = threadIdx.x;
  const int ib = blockIdx.x, jb = blockIdx.y;
  const int i0 = ib * 16, j0 = jb * 16;

  // stage tiles: 16 rows x 32 float4 each (b128 loads, coalesced)
  for (int idx = tid; idx < 16 * 32; idx += 256) {
    const int row = idx >> 5, q = idx & 31;
    sa[row * QROW + q] = A4[(i0 + row) * (ZD / 4) + q];
    sc[row * QROW + q] = C4[(j0 + row) * (ZD / 4) + q];
  }
  if (tid < ZD / 4) sw[tid] = W24[tid];
  __syncthreads();

  const int ti = tid >> 4;
  const int tj = tid & 15;
  const float4* pa = sa + ti * QROW;
  const float4* pc = sc + tj * QROW;

  float s = 0.0f;
#pragma unroll 4
  for (int q = 0; q < ZD / 4; ++q) {
    const float4 xa = pa[q];
    const float4 xc = pc[q];
    const float4 wv = sw[q];
    s = fmaf(fmaxf(xa.x + xc.x, 0.0f), wv.x, s);
    s = fmaf(fmaxf(xa.y + xc.y, 0.0f), wv.y, s);
    s = fmaf(fmaxf(xa.z + xc.z, 0.0f), wv.z, s);
    s = fmaf(fmaxf(xa.w + xc.w, 0.0f), wv.w, s);
  }
  s = fmaxf(s + b2[0], 0.0f);

  ss[ti][tj] = s;
  __syncthreads();

  if (tid < 16) {
    float r = 0.0f;
#pragma unroll
    for (int j = 0; j < 16; ++j) r += ss[tid][j];
    OPpart[jb * NN + i0 + tid] = r;          // unique slot: deterministic
  } else if (tid < 32) {
    const int jj = tid - 16;
    float csum = 0.0f;
#pragma unroll
    for (int i = 0; i < 16; ++i) csum += ss[i][jj];
    COpart[ib * NN + j0 + jj] = csum;        // unique slot: deterministic
  }
}

// ---------------------------------------------------------------------------
// w[i] = sum over 64 block-partials (deterministic replacement for atomics)
// ---------------------------------------------------------------------------
__global__ __launch_bounds__(256) void reduce_parts(const float* __restrict__ part,
                                                    float* __restrict__ w) {
  const int i = blockIdx.x * blockDim.x + threadIdx.x;   // 0..1023
  float s = 0.0f;
#pragma unroll 8
  for (int b = 0; b < NN / 16; ++b) s += part[b * NN + i];
  w[i] = s;
}

// ---------------------------------------------------------------------------
// total = sum(w); probs = w/total (uniform if total == 0); out[h] = probs . X[:,h]
// ---------------------------------------------------------------------------
__global__ __launch_bounds__(256) void finalize_weighted(const float* __restrict__ w,
                                                         const float* __restrict__ X,
                                                         float* __restrict__ out) {
  __shared__ float red[256];
  const int tid = threadIdx.x;

  float local = 0.0f;
  for (int i = tid; i < NN; i += 256) local += w[i];
  red[tid] = local;
  __syncthreads();
  for (int sft = 128; sft > 0; sft >>= 1) {
    if (tid < sft) red[tid] += red[tid + sft];
    __syncthreads();
  }
  const float total = red[0];
  const bool  uni   = !(total > 0.0f);           // scores are >= 0
  const float denom = uni ? (float)NN : total;

  if (tid < ZD) {
    float acc = 0.0f;
    for (int i = 0; i < NN; ++i) {
      const float wi = uni ? 1.0f : w[i];
      acc = fmaf(wi, X[i * ZD + tid], acc);      // column read: coalesced across tid
    }
    out[tid] = acc / denom;
  }
}

// ---------------------------------------------------------------------------
// d_in order: OP_zs[1,1024,128] CO_zs[1,1024,128] W1[256,128] b1[128] W2[128] b2[1]
// d_out: 256 floats = OP_sum[128] ++ CO_sum[128]
// ws (floats): OP_w[1024] | CO_w[1024] | a[1024*128] | c[1024*128]
//              | OPpart[64*1024] | COpart[64*1024]            (~1.6 MB total)
// ---------------------------------------------------------------------------
extern "C" void kernel_launch(void* const* d_in, const int* in_sizes, int n_in,
                              void* d_out, int out_size, void* d_ws, size_t ws_size,
                              hipStream_t stream) {
  const float* OP = (const float*)d_in[0];
  const float* CO = (const float*)d_in[1];
  const float* W1 = (const float*)d_in[2];
  const float* b1 = (const float*)d_in[3];
  const float* W2 = (const float*)d_in[4];
  const float* b2 = (const float*)d_in[5];
  float* out = (float*)d_out;

  float* ws     = (float*)d_ws;
  float* OPw    = ws;
  float* COw    = ws + NN;
  float* Abuf   = ws + 2 * NN;
  float* Cbuf   = Abuf + NN * ZD;
  float* OPpart = Cbuf + NN * ZD;
  float* COpart = OPpart + (NN / 16) * NN;

  // a = op @ W1[:128]          (no bias)
  gemm16_wmma<<<dim3(NN / 16, ZD / 16), dim3(32), 0, stream>>>(OP, W1, nullptr, Abuf);
  // c = co @ W1[128:] + b1     (bias fused)
  gemm16_wmma<<<dim3(NN / 16, ZD / 16), dim3(32), 0, stream>>>(CO, W1 + ZD * ZD, b1, Cbuf);

  pairwise_score<<<dim3(NN / 16, NN / 16), dim3(256), 0, stream>>>(
      (const float4*)Abuf, (const float4*)Cbuf, (const float4*)W2, b2, OPpart, COpart);

  reduce_parts<<<dim3(NN / 256), dim3(256), 0, stream>>>(OPpart, OPw);
  reduce_parts<<<dim3(NN / 256), dim3(256), 0, stream>>>(COpart, COw);

  finalize_weighted<<<dim3(1), dim3(256), 0, stream>>>(OPw, OP, out);
  finalize_weighted<<<dim3(1), dim3(256), 0, stream>>>(COw, CO, out + ZD);
}